// GMELoss3D_31095563223047
// MI455X (gfx1250) — compile-verified
//
#include <hip/hip_runtime.h>
#include <math.h>

#define EPS 1e-6f

typedef float v2f __attribute__((ext_vector_type(2)));
typedef float v8f __attribute__((ext_vector_type(8)));

// Tensor geometry: img (B=2, C=4, H=160, W=160, D=160), D contiguous.
constexpr int NSZ = 160;
constexpr int S_W = 160;
constexpr int S_H = 160 * 160;
constexpr int S_C = 160 * 160 * 160;
constexpr int S_B = 4 * S_C;

// Tiling: per block 8 (W) x 32 (D) outputs, marching a 32-long H chunk.
constexpr int W_T = 8, D_T = 32, H_T = 32;
constexpr int NWT = NSZ / W_T;               // 20
constexpr int NDT = NSZ / D_T;               // 5
constexpr int NHT = NSZ / H_T;               // 5
constexpr int NB  = 2 * NWT * NDT * NHT;     // 1000 blocks

// Reflect index for pad=1 (jnp.pad 'reflect'): -1 -> 1, 160 -> 158.
__device__ __forceinline__ int rid(int i) {
  int j = i < 0 ? -i : i;
  return j >= NSZ ? 2 * (NSZ - 1) - j : j;
}

// Banded Sobel weight matrix W(20x16): rows k = feature (stream a: k=0..8 as
// 3x3 (w,d) taps; stream b: k=9..17; k=18,19 pad=0), cols n = filter (5 used).
// Sx=A_H A_W B_D, Sy=A_H B_W A_D, Sz=B_H A_W A_D,
// Sd11=A_H (U_W B_D - B_W U_D), Sd12=A_H (U_W B_D + B_W U_D).
__device__ __forceinline__ float wmat(int k, int n) {
  if (k >= 18 || n >= 5) return 0.f;
  const float A[3]  = {1.f, 2.f, 1.f};
  const float Bx[3] = {-1.f, 0.f, 1.f};
  int s  = k / 9;
  int jw = (k % 9) / 3;
  int kd = k % 3;
  switch (n) {
    case 0: return (s == 0) ? A[jw] * Bx[kd] : 0.f;
    case 1: return (s == 0) ? Bx[jw] * A[kd] : 0.f;
    case 2: return (s == 1) ? A[jw] * A[kd]  : 0.f;
    case 3: return (s == 0) ? (Bx[kd] - Bx[jw]) : 0.f;
    default:return (s == 0) ? (Bx[kd] + Bx[jw]) : 0.f;
  }
}

// ds_swizzle with immediate pattern; XOR16 swaps the two wave halves.
template <int PAT>
__device__ __forceinline__ float swz(float v) {
  return __int_as_float(__builtin_amdgcn_ds_swizzle(__float_as_int(v), PAT));
}
constexpr int XOR16 = 0x7C10;  // and=0x1f, or=0, xor=16

__global__ __launch_bounds__(256) void gme_main(const float* __restrict__ xin,
                                                const float* __restrict__ yin,
                                                float* __restrict__ part) {
  __shared__ float s_raw[2][2][4][340];  // [input][slot][c][w*34+d] raw planes
  __shared__ float s_cs [2][3][340];     // channel-summed plane ring
  __shared__ float s_ab [2][2][340];     // H-combined a/b planes
  __shared__ float s_g  [8][2][16][8];   // per-wave filter transpose scratch
  __shared__ float s_red[256];

  const int tid  = threadIdx.x;
  const int lane = tid & 31;
  const int wv   = tid >> 5;
  const int m    = lane & 15;
  const int hi   = lane >> 4;

  int bid = blockIdx.x;
  int b   = bid / (NWT * NDT * NHT);
  int r0  = bid % (NWT * NDT * NHT);
  int wt  = r0 / (NDT * NHT);
  int r1  = r0 % (NDT * NHT);
  int dt  = r1 / NHT;
  int ht  = r1 % NHT;
  const int w0 = wt * W_T, d0 = dt * D_T, h0 = ht * H_T;

  // Constant B-matrix (4x16 K x N slices) resident in VGPRs.
  v2f bm[5];
#pragma unroll
  for (int st = 0; st < 5; ++st) {
    int kk = 4 * st + 2 * hi;
    bm[st].x = wmat(kk,     m);
    bm[st].y = wmat(kk + 1, m);
  }

  // Per-lane A-matrix gather offsets into s_ab[inp] (floats).
  const int wr = wv + 1;
  int offA[10];
#pragma unroll
  for (int t = 0; t < 10; ++t) {
    int st = t >> 1;
    int kk = 4 * st + 2 * hi + (t & 1);
    if (kk > 17) kk = 17;              // zero-weight pad: any load is harmless
    int s  = kk / 9;
    int jw = (kk % 9) / 3;
    int kd = kk % 3;
    offA[t] = s * 340 + (wr - 1 + jw) * 34 + (m + kd);
  }

  // Hoisted async-copy address components (h-invariant), byte units for GVS
  // (SGPR base + u32 byte offset).  Plane = 1360 f32 per input -> exactly 6
  // clamp-predicated async loads per thread per input (ASYNCcnt-exact).
  unsigned goffB[6];
  unsigned ldsrel[6];
#pragma unroll
  for (int i = 0; i < 6; ++i) {
    int e = tid + 256 * i;
    if (e > 1359) e = 1359;
    int c = e / 340, pos = e % 340, wrr = pos / 34, dc = pos % 34;
    goffB[i] = (unsigned)(b * S_B + c * S_C + rid(w0 - 1 + wrr) * S_W +
                          rid(d0 - 1 + dc)) * 4u;
    ldsrel[i] = (unsigned)((c * 340 + pos) * 4);
  }
  const unsigned rawb =
      (unsigned)(unsigned long long)(&s_raw[0][0][0][0]);  // low 32b = LDS addr

  // Hoisted stage-task decomposition (680 csum/ab element tasks).
  int  tinp[3], tpos[3];
  bool tval[3];
#pragma unroll
  for (int j = 0; j < 3; ++j) {
    int t = tid + 256 * j;
    tval[j] = t < 680;
    int tt  = tval[j] ? t : 0;
    tinp[j] = tt / 340;
    tpos[j] = tt % 340;
  }

  // Async plane fetch into raw slot (p+1)&1: 12 async insts per wave (GVS).
  auto load_plane = [&](int p) {
    int slot = (p + 1) & 1;
    unsigned hoffB = (unsigned)(rid(p) * S_H) * 4u;
#pragma unroll
    for (int inp = 0; inp < 2; ++inp) {
      const float* src = inp ? yin : xin;
      unsigned lb = rawb + (unsigned)(inp * 10880 + slot * 5440);
#pragma unroll
      for (int i = 0; i < 6; ++i) {
        unsigned vo = goffB[i] + hoffB;
        unsigned lo = lb + ldsrel[i];
        asm volatile("global_load_async_to_lds_b32 %0, %1, %2"
                     :: "v"(lo), "v"(vo), "s"(src) : "memory");
      }
    }
  };
  // Channel-sum plane p: raw slot (p+1)&1 -> cs slot (p+1)%3.
  auto csum_plane = [&](int p) {
    int slot = (p + 1) & 1, cslot = (p + 1) % 3;
#pragma unroll
    for (int j = 0; j < 3; ++j)
      if (tval[j]) {
        const float* rp = &s_raw[tinp[j]][slot][0][tpos[j]];
        s_cs[tinp[j]][cslot][tpos[j]] = (rp[0] + rp[340]) + (rp[680] + rp[1020]);
      }
  };

  // Prologue: planes h0-1, h0; build their csums; then launch h0+1.
  load_plane(h0 - 1);
  load_plane(h0);
  asm volatile("s_wait_asynccnt 0" ::: "memory");
  __syncthreads();
  csum_plane(h0 - 1);
  csum_plane(h0);
  __syncthreads();
  load_plane(h0 + 1);

  float acc = 0.f;
  for (int h = h0; h < h0 + H_T; ++h) {
    // Prefetch plane h+2; wait so that h+1 (issued last iter) has landed.
    load_plane(h + 2);
    asm volatile("s_wait_asynccnt 12" ::: "memory");
    __syncthreads();

    csum_plane(h + 1);
    __syncthreads();

    // H-axis combine: a = c[-1]+2c[0]+c[+1], b = c[+1]-c[-1].
    {
      int s0 = h % 3, s1 = (h + 1) % 3, s2 = (h + 2) % 3;
#pragma unroll
      for (int j = 0; j < 3; ++j)
        if (tval[j]) {
          float cm1 = s_cs[tinp[j]][s0][tpos[j]];
          float c0  = s_cs[tinp[j]][s1][tpos[j]];
          float cp1 = s_cs[tinp[j]][s2][tpos[j]];
          s_ab[tinp[j]][0][tpos[j]] = cm1 + 2.f * c0 + cp1;
          s_ab[tinp[j]][1][tpos[j]] = cp1 - cm1;
        }
    }
    __syncthreads();

    // Stage B: per wave, two 16-voxel D chunks.  Preload ALL 10 A-fragments
    // (2 inputs x 5 K-steps) into distinct registers so the DS loads form one
    // clause covered by a single wait, then run the 10
    // v_wmma_f32_16x16x4_f32 as two interleaved accumulator chains.
#pragma unroll
    for (int ch = 0; ch < 2; ++ch) {
      int dbase = 16 * ch;
      v2f av[2][5];
#pragma unroll
      for (int inp = 0; inp < 2; ++inp) {
        const float* abp = &s_ab[inp][0][0];
#pragma unroll
        for (int st = 0; st < 5; ++st) {
          av[inp][st].x = abp[offA[2 * st]     + dbase];
          av[inp][st].y = abp[offA[2 * st + 1] + dbase];
        }
      }
      v8f cx = {}, cy = {};
#pragma unroll
      for (int st = 0; st < 5; ++st) {
        cx = __builtin_amdgcn_wmma_f32_16x16x4_f32(
            false, av[0][st], false, bm[st], (short)0, cx, false, false);
        cy = __builtin_amdgcn_wmma_f32_16x16x4_f32(
            false, av[1][st], false, bm[st], (short)0, cy, false, false);
      }
      // Micro-transpose: lane n (n<5) parks filter column n in per-wave LDS
      // scratch (imm-offset paired stores; wave-internal DS stays in order).
      if (m < 5) {
        float* spx = &s_g[wv][0][8 * hi][m];
        float* spy = &s_g[wv][1][8 * hi][m];
#pragma unroll
        for (int r = 0; r < 8; ++r) {
          spx[r * 8] = cx[r];
          spy[r * 8] = cy[r];
        }
      }
      // Re-load along the filter axis: lanes 0-15 -> input x voxel m,
      // lanes 16-31 -> input y voxel m.  One b128 + one b32 per lane.
      const float* lp = &s_g[wv][hi][m][0];
      float s = 0.f;
#pragma unroll
      for (int j = 0; j < 5; ++j) {
        float q = lp[j] + 4.f * EPS;
        s += q * q + EPS;
      }
      float e = sqrtf(s);
      // xor16 pairs ex with ey; both halves add the same |diff| -> 2x,
      // compensated in the 0.125 = (1/C)/2 factor.  Branchless, exec-full.
      acc += 0.125f * fabsf(e - swz<XOR16>(e));
    }
  }

  // Deterministic block reduction -> per-block partial.
  s_red[tid] = acc;
  __syncthreads();
  for (int s = 128; s > 0; s >>= 1) {
    if (tid < s) s_red[tid] += s_red[tid + s];
    __syncthreads();
  }
  if (tid == 0) part[blockIdx.x] = s_red[0];
}

__global__ __launch_bounds__(256) void gme_finalize(const float* __restrict__ part,
                                                    float* __restrict__ out) {
  __shared__ float red[256];
  float s = 0.f;
  for (int i = threadIdx.x; i < NB; i += 256) s += part[i];  // fixed order
  red[threadIdx.x] = s;
  __syncthreads();
  for (int sft = 128; sft > 0; sft >>= 1) {
    if (threadIdx.x < sft) red[threadIdx.x] += red[threadIdx.x + sft];
    __syncthreads();
  }
  if (threadIdx.x == 0)
    out[0] = EPS + red[0] / 8192000.f;  // mean over B*H*W*D = 2*160^3
}

extern "C" void kernel_launch(void* const* d_in, const int* in_sizes, int n_in,
                              void* d_out, int out_size, void* d_ws, size_t ws_size,
                              hipStream_t stream) {
  (void)in_sizes; (void)n_in; (void)out_size; (void)ws_size;
  const float* x = (const float*)d_in[0];
  const float* y = (const float*)d_in[1];
  float* part = (float*)d_ws;   // NB floats of scratch
  float* out  = (float*)d_out;
  gme_main<<<NB, 256, 0, stream>>>(x, y, part);
  gme_finalize<<<1, 256, 0, stream>>>(part, out);
}